// HadamardTransform_3745211483014
// MI455X (gfx1250) — compile-verified
//
#include <hip/hip_runtime.h>

// FWHT-128 over contiguous groups, H128 = H8 (rows) ⊗ H16 (cols).
// Two groups per 16x16 f32 tile; stage 1 = V_WMMA_F32_16X16X4_F32 x4 (batched
// b64 loads, accumulator chain), stage 2 = per-lane register butterflies.

typedef __attribute__((ext_vector_type(2))) float v2f;
typedef __attribute__((ext_vector_type(8))) float v8f;

#define TILES_PER_WAVE 4
#define WAVES_PER_BLOCK 8

__device__ __forceinline__ void fwht8_regs(v8f& c) {
    // bit 0
    float b0 = c[0] + c[1], b1 = c[0] - c[1];
    float b2 = c[2] + c[3], b3 = c[2] - c[3];
    float b4 = c[4] + c[5], b5 = c[4] - c[5];
    float b6 = c[6] + c[7], b7 = c[6] - c[7];
    // bit 1
    float d0 = b0 + b2, d2 = b0 - b2;
    float d1 = b1 + b3, d3 = b1 - b3;
    float d4 = b4 + b6, d6 = b4 - b6;
    float d5 = b5 + b7, d7 = b5 - b7;
    // bit 2
    c[0] = d0 + d4; c[4] = d0 - d4;
    c[1] = d1 + d5; c[5] = d1 - d5;
    c[2] = d2 + d6; c[6] = d2 - d6;
    c[3] = d3 + d7; c[7] = d3 - d7;
}

// base: float offset of this tile; aoff = row*16 + 2*half (A-layout lane offset);
// soff = 8*half*16 + row (D-layout lane offset for stores).
__device__ __forceinline__ void do_tile(const float* __restrict__ x,
                                        float* __restrict__ y,
                                        unsigned base, unsigned aoff, unsigned soff,
                                        const v2f* B) {
    const unsigned a0 = base + aoff;
    // Batch all four K-chunk loads so they overlap in flight (one clause).
    v2f a[4];
    a[0] = *(const v2f*)(x + a0);
    a[1] = *(const v2f*)(x + a0 + 4);
    a[2] = *(const v2f*)(x + a0 + 8);
    a[3] = *(const v2f*)(x + a0 + 12);

    v8f c = {0.f, 0.f, 0.f, 0.f, 0.f, 0.f, 0.f, 0.f};
    #pragma unroll
    for (int kc = 0; kc < 4; ++kc) {
        c = __builtin_amdgcn_wmma_f32_16x16x4_f32(
                /*neg_a=*/false, a[kc], /*neg_b=*/false, B[kc],
                /*c_mod=*/(short)0, c, /*reuse_a=*/false, /*reuse_b=*/false);
    }
    // Stage 2: H8 over rows — rows of each group live in this lane's 8 acc regs.
    fwht8_regs(c);
    // D-layout store: vgpr j holds element (m = j + 8*half, n = row).
    #pragma unroll
    for (int j = 0; j < 8; ++j) {
        y[base + soff + (unsigned)j * 16] = c[j];
    }
}

__global__ __launch_bounds__(32 * WAVES_PER_BLOCK)
void hadamard128_wmma_kernel(const float* __restrict__ x,
                             float* __restrict__ y,
                             unsigned numTiles) {
    const unsigned lane = threadIdx.x & 31u;
    const unsigned row  = lane & 15u;  // N index / M index for A loads
    const unsigned half = lane >> 4;   // lane half selects K sub-pair / upper rows

    // Stage-1 operand: B[kc][v][lane] = H16[K][N] * (1/sqrt(128)),
    // K = 4*kc + 2*half + v, N = row. Generated in registers, reused all tiles.
    const float s = 0.08838834764831845f; // 1/sqrt(128)
    v2f B[4];
    #pragma unroll
    for (int kc = 0; kc < 4; ++kc) {
        unsigned K0 = 4u * kc + 2u * half;
        unsigned K1 = K0 + 1u;
        B[kc].x = (__popc(K0 & row) & 1) ? -s : s;
        B[kc].y = (__popc(K1 & row) & 1) ? -s : s;
    }

    const unsigned aoff = row * 16u + 2u * half;      // A-layout lane offset
    const unsigned soff = half * 128u + row;          // D-layout lane offset

    const unsigned wave  = blockIdx.x * WAVES_PER_BLOCK + (threadIdx.x >> 5);
    const unsigned tile0 = wave * TILES_PER_WAVE;

    if (tile0 + TILES_PER_WAVE <= numTiles) {
        const unsigned base0 = tile0 * 256u;
        // Hint the next wave-chunk into cache (emits global_prefetch_b8).
        __builtin_prefetch(x + base0 + TILES_PER_WAVE * 256u, 0, 1);
        #pragma unroll
        for (int t = 0; t < TILES_PER_WAVE; ++t) {
            do_tile(x, y, base0 + (unsigned)t * 256u, aoff, soff, B);
        }
    } else {
        for (unsigned tile = tile0; tile < numTiles; ++tile) {
            do_tile(x, y, tile * 256u, aoff, soff, B);
        }
    }
}

extern "C" void kernel_launch(void* const* d_in, const int* in_sizes, int n_in,
                              void* d_out, int out_size, void* d_ws, size_t ws_size,
                              hipStream_t stream) {
    (void)n_in; (void)d_ws; (void)ws_size; (void)out_size;
    const float* x = (const float*)d_in[0];
    float*       y = (float*)d_out;

    const long n = (long)in_sizes[0];               // 8192*8192, multiple of 256
    const unsigned numTiles = (unsigned)(n / 256);  // two 128-groups per tile
    const unsigned waves  = (numTiles + TILES_PER_WAVE - 1) / TILES_PER_WAVE;
    const unsigned blocks = (waves + WAVES_PER_BLOCK - 1) / WAVES_PER_BLOCK;

    hadamard128_wmma_kernel<<<blocks, 32 * WAVES_PER_BLOCK, 0, stream>>>(x, y, numTiles);
}